// PlannerHead_31610959298858
// MI455X (gfx1250) — compile-verified
//
#include <hip/hip_runtime.h>

typedef __attribute__((ext_vector_type(2))) float v2f;
typedef __attribute__((ext_vector_type(8))) float v8f;
typedef __attribute__((ext_vector_type(2))) int   i32x2;
typedef __attribute__((ext_vector_type(4))) int   i32x4;

#define HDIM   1024
#define NSLOTS 8
#define VOCAB  32768
#define BATCH  32
#define TSEQ   2048
#define BSROWS (BATCH * NSLOTS)   // 256

// ---- gfx1250 async global->LDS path (guarded so the file always compiles) --
#if defined(__AMDGCN__) &&                                            \
    __has_builtin(__builtin_amdgcn_global_load_async_to_lds_b128) &&  \
    __has_builtin(__builtin_amdgcn_global_load_async_to_lds_b64) &&   \
    __has_builtin(__builtin_amdgcn_s_wait_asynccnt)
#define USE_ASYNC_LDS 1
#else
#define USE_ASYNC_LDS 0
#endif

#define GLOBAL_AS __attribute__((address_space(1)))
#define LDS_AS    __attribute__((address_space(3)))

// ---------------------------------------------------------------------------
// Kernel 1: masked mean pooling.  grid = B, block = 256 (each thread = float4)
// ---------------------------------------------------------------------------
__global__ __launch_bounds__(256)
void k_pool(const float* __restrict__ hs, const int* __restrict__ mask,
            float* __restrict__ pooled) {
  __shared__ float red[256];
  const int b = blockIdx.x, tid = threadIdx.x;

  float ms = 0.f;
  for (int t = tid; t < TSEQ; t += 256) ms += (float)mask[b * TSEQ + t];
  red[tid] = ms; __syncthreads();
  for (int s = 128; s > 0; s >>= 1) { if (tid < s) red[tid] += red[tid + s]; __syncthreads(); }
  const float rden = 1.f / fmaxf(red[0], 1.f);
  __syncthreads();

  float4 acc = {0.f, 0.f, 0.f, 0.f};
  const size_t base = (size_t)b * TSEQ * HDIM;
  for (int t = 0; t < TSEQ; ++t) {
    const float w = (float)mask[b * TSEQ + t];
    const float4 v = ((const float4*)(hs + base + (size_t)t * HDIM))[tid];
    acc.x += w * v.x; acc.y += w * v.y; acc.z += w * v.z; acc.w += w * v.w;
  }
  acc.x *= rden; acc.y *= rden; acc.z *= rden; acc.w *= rden;
  ((float4*)(pooled + (size_t)b * HDIM))[tid] = acc;
}

// ---------------------------------------------------------------------------
// Kernel 2: squared row norms of a [rows, 1024] matrix. grid = rows, block=256
// ---------------------------------------------------------------------------
__global__ __launch_bounds__(256)
void k_rownorm(const float* __restrict__ M, float* __restrict__ out) {
  __shared__ float red[256];
  const int r = blockIdx.x, tid = threadIdx.x;
  const float4 v = ((const float4*)(M + (size_t)r * HDIM))[tid];
  red[tid] = v.x * v.x + v.y * v.y + v.z * v.z + v.w * v.w;
  __syncthreads();
  for (int s = 128; s > 0; s >>= 1) { if (tid < s) red[tid] += red[tid + s]; __syncthreads(); }
  if (tid == 0) out[r] = red[0];
}

// ---------------------------------------------------------------------------
// Kernel 3: WMMA GEMM  D = A[M,K] * B[N,K]^T  (fp32, V_WMMA_F32_16X16X4_F32)
//   grid.x = N/128, grid.y = M/16; block = 256 threads = 8 waves.
//   Double-buffered K-chunks of 32 staged via GLOBAL_LOAD_ASYNC_TO_LDS
//   (ASYNCcnt-tracked, no VGPR round trip) so chunk c+1 streams in while the
//   WMMAs consume chunk c.  Row stride 36 dwords -> conflict-free frag reads.
//   mode 0: D[m][n] = acc                       (projection; mirror copy D2)
//   mode 1: D[m][n] = 2*acc - qn[m] - cn[n]     (negative squared distance)
// ---------------------------------------------------------------------------
#define KC   32
#define LDSP 36          // 32 + 4 pad dwords per row; 36 = 4*9, 9 coprime 16

// Issue one K-chunk's staging: per thread 1 x B64 (A) + 4 x B128 (B) = 5
// async instructions per wave (uniform across waves; EXEC all ones).
__device__ __forceinline__
void stage_chunk(const float* __restrict__ A, const float* __restrict__ Bm,
                 int K, int m0, int n0, int kc,
                 float* __restrict__ as, float* __restrict__ bs, int tid) {
  // A chunk: 16 rows x 32 cols = 256 float2, one per thread
  {
    const int row = tid >> 4;          // 0..15
    const int c2  = tid & 15;          // 0..15 -> col = c2*2
    const float* g = A + (size_t)(m0 + row) * K + kc + c2 * 2;
    float* l = &as[row * LDSP + c2 * 2];
#if USE_ASYNC_LDS
    __builtin_amdgcn_global_load_async_to_lds_b64(
        (GLOBAL_AS i32x2*)g, (LDS_AS i32x2*)l, 0, 0);
#else
    const float2 v = *(const float2*)g;
    l[0] = v.x; l[1] = v.y;
#endif
  }
  // B chunk: 128 rows x 32 cols = 1024 float4, four per thread
  #pragma unroll
  for (int j = 0; j < 4; ++j) {
    const int idx = j * 256 + tid;     // 0..1023
    const int row = idx >> 3;          // 0..127  (8 float4 per row)
    const int c4  = idx & 7;           // 0..7
    const float* g = Bm + (size_t)(n0 + row) * K + kc + c4 * 4;
    float* l = &bs[row * LDSP + c4 * 4];
#if USE_ASYNC_LDS
    __builtin_amdgcn_global_load_async_to_lds_b128(
        (GLOBAL_AS i32x4*)g, (LDS_AS i32x4*)l, 0, 0);
#else
    const float4 v = *(const float4*)g;
    l[0] = v.x; l[1] = v.y; l[2] = v.z; l[3] = v.w;
#endif
  }
}

__global__ __launch_bounds__(256)
void k_gemm_nt(const float* __restrict__ A, const float* __restrict__ Bm,
               int K, float* __restrict__ D, long ldD, float* __restrict__ D2,
               const float* __restrict__ qn, const float* __restrict__ cn,
               int mode) {
  __shared__ float As[2][16 * LDSP];    //  2 x 2,304 B
  __shared__ float Bs[2][128 * LDSP];   //  2 x 18,432 B  (total ~41.5 KB)

  const int tid  = threadIdx.x;
  const int wave = tid >> 5;
  const int lane = tid & 31;
  const int half = lane >> 4;
  const int l16  = lane & 15;
  const int m0 = blockIdx.y * 16;
  const int n0 = blockIdx.x * 128;
  const int nl = wave * 16 + l16;

  const int nchunk = K / KC;
  stage_chunk(A, Bm, K, m0, n0, 0, As[0], Bs[0], tid);

  v8f acc = {};
  for (int c = 0; c < nchunk; ++c) {
    const int cur = c & 1;
    if (c + 1 < nchunk) {
      stage_chunk(A, Bm, K, m0, n0, (c + 1) * KC, As[cur ^ 1], Bs[cur ^ 1], tid);
#if USE_ASYNC_LDS
      __builtin_amdgcn_s_wait_asynccnt(5);  // chunk c done; c+1 still in flight
#endif
    } else {
#if USE_ASYNC_LDS
      __builtin_amdgcn_s_wait_asynccnt(0);
#endif
    }
    __syncthreads();   // all waves' chunk-c data visible in LDS

    const float* as = As[cur];
    const float* bs = Bs[cur];
    #pragma unroll
    for (int kk = 0; kk < KC; kk += 4) {
      // A 16x4 frag: lanes 0-15 hold K=kk,kk+1; lanes 16-31 hold K=kk+2,kk+3
      const v2f a = *(const v2f*)&as[l16 * LDSP + kk + 2 * half];
      // B 4x16 frag: same K striping, lanes index N
      const v2f b = *(const v2f*)&bs[nl * LDSP + kk + 2 * half];
      acc = __builtin_amdgcn_wmma_f32_16x16x4_f32(
                false, a, false, b, (short)0, acc, false, false);
    }
    __syncthreads();   // done reading buf before it is re-staged in c+2
  }

  // epilogue: lane l, VGPR r -> M = r + 8*(l>>4), N = l&15
  const int nG = n0 + wave * 16 + l16;
  #pragma unroll
  for (int r = 0; r < 8; ++r) {
    const int mG = m0 + r + 8 * half;
    float v = acc[r];
    if (mode == 1) v = 2.f * v - qn[mG] - cn[nG];
    D[(size_t)mG * ldD + nG] = v;
    if (D2) D2[(size_t)mG * ldD + nG] = v;
  }
}

// ---------------------------------------------------------------------------
// Kernel 4: argmax over V per (b,s) row.  grid = 256, block = 256.
// First-index tie-break to match jnp.argmax.
// ---------------------------------------------------------------------------
__global__ __launch_bounds__(256)
void k_argmax(const float* __restrict__ logits, int* __restrict__ idxOut,
              float* __restrict__ idxF) {
  __shared__ float bm[256];
  __shared__ int   bi[256];
  const int row = blockIdx.x, tid = threadIdx.x;
  const float* L = logits + (size_t)row * VOCAB;
  float best = -3.402823466e38f; int bidx = 0;
  for (int v = tid; v < VOCAB; v += 256) {
    const float x = L[v];
    if (x > best) { best = x; bidx = v; }
  }
  bm[tid] = best; bi[tid] = bidx; __syncthreads();
  for (int s = 128; s > 0; s >>= 1) {
    if (tid < s) {
      if (bm[tid + s] > bm[tid] ||
          (bm[tid + s] == bm[tid] && bi[tid + s] < bi[tid])) {
        bm[tid] = bm[tid + s]; bi[tid] = bi[tid + s];
      }
    }
    __syncthreads();
  }
  if (tid == 0) { idxOut[row] = bi[0]; idxF[row] = (float)bi[0]; }
}

// ---------------------------------------------------------------------------
// Kernel 5: gather embedded -> quantized, per-row squared-diff partial sums.
// ---------------------------------------------------------------------------
__global__ __launch_bounds__(256)
void k_embed(const float* __restrict__ codebook, const int* __restrict__ idx,
             const float* __restrict__ preq, float* __restrict__ quantOut,
             float* __restrict__ rowloss) {
  __shared__ float red[256];
  const int row = blockIdx.x, tid = threadIdx.x;
  const int code = idx[row];
  const float4 e = ((const float4*)(codebook + (size_t)code * HDIM))[tid];
  const float4 p = ((const float4*)(preq + (size_t)row * HDIM))[tid];
  ((float4*)(quantOut + (size_t)row * HDIM))[tid] = e;
  const float dx = p.x - e.x, dy = p.y - e.y, dz = p.z - e.z, dw = p.w - e.w;
  red[tid] = dx * dx + dy * dy + dz * dz + dw * dw;
  __syncthreads();
  for (int s = 128; s > 0; s >>= 1) { if (tid < s) red[tid] += red[tid + s]; __syncthreads(); }
  if (tid == 0) rowloss[row] = red[0];
}

// ---------------------------------------------------------------------------
// Kernel 6: deterministic loss finalize (no float atomics).
// ---------------------------------------------------------------------------
__global__ __launch_bounds__(256)
void k_finalize(const float* __restrict__ rowloss, float* __restrict__ outLoss) {
  __shared__ float red[256];
  const int tid = threadIdx.x;
  red[tid] = rowloss[tid];
  __syncthreads();
  for (int s = 128; s > 0; s >>= 1) { if (tid < s) red[tid] += red[tid + s]; __syncthreads(); }
  if (tid == 0) {
    const float m = red[0] / (float)(BSROWS * HDIM);
    outLoss[0] = m;   // commitment_loss
    outLoss[1] = m;   // codebook_loss (identical in forward)
  }
}

// ---------------------------------------------------------------------------
extern "C" void kernel_launch(void* const* d_in, const int* in_sizes, int n_in,
                              void* d_out, int out_size, void* d_ws, size_t ws_size,
                              hipStream_t stream) {
  const float* hs       = (const float*)d_in[0];  // [32,2048,1024]
  const int*   mask     = (const int*)  d_in[1];  // [32,2048]
  const float* Wproj    = (const float*)d_in[2];  // [8192,1024]
  const float* codebook = (const float*)d_in[3];  // [32768,1024]

  float* out = (float*)d_out;
  // d_out layout (flat, return order):
  float* out_logits = out;                                     // 256*32768
  float* out_idxF   = out_logits + (size_t)BSROWS * VOCAB;     // 256
  float* out_preq   = out_idxF + BSROWS;                       // 256*1024
  float* out_quant  = out_preq + (size_t)BSROWS * HDIM;        // 256*1024
  float* out_loss   = out_quant + (size_t)BSROWS * HDIM;       // 2

  // workspace layout
  float* ws      = (float*)d_ws;
  float* pooled  = ws;                               // 32*1024
  float* preq_ws = pooled + BATCH * HDIM;            // 256*1024
  float* qnorm   = preq_ws + (size_t)BSROWS * HDIM;  // 256
  float* cnorm   = qnorm + BSROWS;                   // 32768
  float* rowloss = cnorm + VOCAB;                    // 256
  int*   idx_ws  = (int*)(rowloss + BSROWS);         // 256

  // 1) masked mean pool: [32,2048,1024] -> [32,1024]
  k_pool<<<BATCH, 256, 0, stream>>>(hs, mask, pooled);

  // 2) codebook squared norms (independent of pooling)
  k_rownorm<<<VOCAB, 256, 0, stream>>>(codebook, cnorm);

  // 3) pre_q = pooled @ W_proj^T : M=32, N=8192, K=1024
  k_gemm_nt<<<dim3(NSLOTS * HDIM / 128, BATCH / 16), 256, 0, stream>>>(
      pooled, Wproj, HDIM, out_preq, (long)(NSLOTS * HDIM), preq_ws,
      nullptr, nullptr, 0);

  // 4) pre_q squared row norms
  k_rownorm<<<BSROWS, 256, 0, stream>>>(preq_ws, qnorm);

  // 5) logits = 2 q.c - |q|^2 - |c|^2 : M=256, N=32768, K=1024
  k_gemm_nt<<<dim3(VOCAB / 128, BSROWS / 16), 256, 0, stream>>>(
      preq_ws, codebook, HDIM, out_logits, (long)VOCAB, nullptr,
      qnorm, cnorm, 1);

  // 6) argmax per row (first-max tie-break)
  k_argmax<<<BSROWS, 256, 0, stream>>>(out_logits, idx_ws, out_idxF);

  // 7) quantized = codebook[idx]; per-row loss partials
  k_embed<<<BSROWS, 256, 0, stream>>>(codebook, idx_ws, preq_ws, out_quant, rowloss);

  // 8) deterministic scalar losses
  k_finalize<<<1, 256, 0, stream>>>(rowloss, out_loss);
}